// FixedMaskedDendriteLayer_40175124086887
// MI455X (gfx1250) — compile-verified
//
#include <hip/hip_runtime.h>

typedef __attribute__((ext_vector_type(2))) float v2f;
typedef __attribute__((ext_vector_type(2))) float f32x2;
typedef __attribute__((ext_vector_type(8))) float v8f;

#define BATCH   256
#define IN_DIM  4096
#define SOMA    2048
#define BRANCH  16
#define ND      32768      // SOMA * BRANCH
#define SS      32         // samples per dendrite
#define BT      16         // batch tile (== WMMA M)
#define STL     32         // soma tile (2 WMMA groups of 16)
#define DT      (STL*BRANCH)  // 512 dendrites per block
#define THREADS 512
#define XSTRIDE 18         // padded row stride (floats) for transposed x in LDS
#define NEG_SLOPE 0.1f

#if __has_builtin(__builtin_amdgcn_global_load_async_to_lds_b32) && \
    __has_builtin(__builtin_amdgcn_s_wait_asynccnt)
#define USE_ASYNC_LDS 1
#endif

__device__ __forceinline__ float leaky(float v) { return v >= 0.f ? v : NEG_SLOPE * v; }

extern "C" __global__ __launch_bounds__(THREADS)
void dendrite_soma_kernel(const float* __restrict__ x,
                          const int*   __restrict__ idx,
                          const float* __restrict__ sw,
                          const float* __restrict__ sb,
                          const float* __restrict__ cw,
                          const float* __restrict__ somab,
                          float* __restrict__ out_soma,
                          float* __restrict__ out_dend)
{
    extern __shared__ float smem[];
    float* xs = smem;                       // transposed x: [j][b], 4096 * 18 floats (288 KB)
    // acts aliases the start of xs after phase 2 (xs dead by then): 2*256*18 floats (36 KB)
    float* acts = smem;

    const int t  = threadIdx.x;
    const int bt = blockIdx.x;   // batch tile index
    const int st = blockIdx.y;   // soma tile index

    // -------- Phase 1: stage 16 x-rows into LDS, TRANSPOSED xs[j*18 + b] ----
    {
        const float* gsrc = x + (size_t)bt * BT * IN_DIM;
#ifdef USE_ASYNC_LDS
        typedef __attribute__((address_space(1))) int* gp_t;
        typedef __attribute__((address_space(3))) int* lp_t;
        #pragma unroll 8
        for (int i = 0; i < (BT * IN_DIM) / THREADS; ++i) {   // 128 elems/thread
            const int e = t + i * THREADS;                    // coalesced global
            const int b = e >> 12, j = e & 4095;
            __builtin_amdgcn_global_load_async_to_lds_b32(
                (gp_t)(gsrc + e), (lp_t)(xs + j * XSTRIDE + b), 0, 0);
        }
        __builtin_amdgcn_s_wait_asynccnt(0);
#else
        const float4* xsrc = (const float4*)gsrc;
        #pragma unroll 4
        for (int i = 0; i < (BT * IN_DIM / 4) / THREADS; ++i) {  // 32 float4/thread
            const int e4 = t + i * THREADS;
            const float4 v = xsrc[e4];
            const int b = e4 >> 10, j = (e4 & 1023) * 4;
            float* dst = xs + j * XSTRIDE + b;
            dst[0]           = v.x;
            dst[XSTRIDE]     = v.y;
            dst[2 * XSTRIDE] = v.z;
            dst[3 * XSTRIDE] = v.w;
        }
#endif
    }
    __syncthreads();

    // -------- Phase 2: one dendrite per thread, 16 batches (8 f32x2 pairs) --
    const int d = st * DT + t;
    f32x2 acc[BT / 2];
    {
        const float bias = sb[d];
        #pragma unroll
        for (int p = 0; p < BT / 2; ++p) { acc[p].x = bias; acc[p].y = bias; }
    }
    const int4*   irow = (const int4*)(idx + (size_t)d * SS);
    const float4* wrow = (const float4*)(sw  + (size_t)d * SS);
    __builtin_prefetch(irow, 0, 1);   // global_prefetch_b8
    __builtin_prefetch(wrow, 0, 1);

    #pragma unroll
    for (int s4 = 0; s4 < SS / 4; ++s4) {
        const int4   i4 = irow[s4];
        const float4 w4 = wrow[s4];
        const f32x2* p0 = (const f32x2*)(xs + i4.x * XSTRIDE);  // 72B row, 8B aligned
        const f32x2* p1 = (const f32x2*)(xs + i4.y * XSTRIDE);
        const f32x2* p2 = (const f32x2*)(xs + i4.z * XSTRIDE);
        const f32x2* p3 = (const f32x2*)(xs + i4.w * XSTRIDE);
        #pragma unroll
        for (int p = 0; p < BT / 2; ++p) acc[p] += p0[p] * w4.x;  // ds_load_b64 + v_pk_fma_f32
        #pragma unroll
        for (int p = 0; p < BT / 2; ++p) acc[p] += p1[p] * w4.y;
        #pragma unroll
        for (int p = 0; p < BT / 2; ++p) acc[p] += p2[p] * w4.z;
        #pragma unroll
        for (int p = 0; p < BT / 2; ++p) acc[p] += p3[p] * w4.w;
    }

    // leaky-ReLU + coalesced dendritic_act stores (lanes = consecutive d)
    float act_v[BT];
    #pragma unroll
    for (int p = 0; p < BT / 2; ++p) {
        act_v[2 * p]     = leaky(acc[p].x);
        act_v[2 * p + 1] = leaky(acc[p].y);
    }
    #pragma unroll
    for (int b = 0; b < BT; ++b)
        out_dend[(size_t)(bt * BT + b) * ND + d] = act_v[b];

    __syncthreads();   // all reads of xs complete -> safe to alias acts over it

    // stage activations for WMMA: acts[group][k'][b], padded row stride 18
    {
        const int g  = t >> 8;    // 16-soma group (0/1)
        const int kp = t & 255;   // k' = local_soma*16 + branch
        float* arow = acts + (g * 256 + kp) * XSTRIDE;
        #pragma unroll
        for (int b = 0; b < BT; ++b) arow[b] = act_v[b];
    }
    __syncthreads();

    // -------- Phase 3: soma reduction as exact GEMM via f32 WMMA ------------
    // D[b,n'] = sum_{k'=0..255} A[b,k'] * B[k',n'],  B block-diagonal cable wts
    if (t < 64) {                       // waves 0,1 (full EXEC per wave)
        const int lane  = t & 31;
        const int g2    = t >> 5;       // group handled by this wave
        const int m     = lane & 15;    // batch row (A: M = lane)
        const int koff  = (lane < 16) ? 0 : 2;   // A/B K sub-lanes per ISA layout
        const int ncol  = lane & 15;    // output column N = lane
        const int nglob = st * STL + g2 * 16 + ncol;

        float cwreg[16];
        {
            const float4* cwp = (const float4*)(cw + (size_t)nglob * BRANCH);
            #pragma unroll
            for (int q = 0; q < 4; ++q) {
                const float4 v = cwp[q];
                cwreg[4*q+0] = v.x; cwreg[4*q+1] = v.y;
                cwreg[4*q+2] = v.z; cwreg[4*q+3] = v.w;
            }
        }

        const float* ag = acts + g2 * 256 * XSTRIDE;
        v8f c = {0.f, 0.f, 0.f, 0.f, 0.f, 0.f, 0.f, 0.f};

        #pragma unroll 4
        for (int c4 = 0; c4 < 64; ++c4) {
            const int k0 = 4 * c4 + koff;
            v2f a, bm;
            a.x = ag[k0 * XSTRIDE + m];
            a.y = ag[(k0 + 1) * XSTRIDE + m];
            // only column k'>>4 (== c4>>2) of B is nonzero in this K-chunk
            const bool on = (ncol == (c4 >> 2));
            bm.x = on ? cwreg[k0 & 15]       : 0.f;
            bm.y = on ? cwreg[(k0 + 1) & 15] : 0.f;
            c = __builtin_amdgcn_wmma_f32_16x16x4_f32(
                    /*neg_a=*/false, a, /*neg_b=*/false, bm,
                    /*c_mod=*/(short)0, c, /*reuse_a=*/false, /*reuse_b=*/false);
        }

        const float bias = somab[nglob];
        #pragma unroll
        for (int j = 0; j < 8; ++j) {          // D: VGPR j = rows j / j+8
            const int brow = j + ((lane < 16) ? 0 : 8);
            const float v = leaky(c[j] + bias);
            out_soma[(size_t)(bt * BT + brow) * SOMA + nglob] = v;
        }
    }
}

extern "C" void kernel_launch(void* const* d_in, const int* in_sizes, int n_in,
                              void* d_out, int out_size, void* d_ws, size_t ws_size,
                              hipStream_t stream)
{
    const float* x     = (const float*)d_in[0];
    const int*   idx   = (const int*)  d_in[1];
    const float* sw    = (const float*)d_in[2];
    const float* sb    = (const float*)d_in[3];
    const float* cwp   = (const float*)d_in[4];
    const float* somab = (const float*)d_in[5];

    float* out_soma = (float*)d_out;                                  // 256*2048
    float* out_dend = (float*)d_out + (size_t)BATCH * SOMA;           // 256*32768

    dim3 grid(BATCH / BT, SOMA / STL);   // 16 x 64 workgroups
    const size_t shmem = (size_t)(IN_DIM * XSTRIDE) * sizeof(float);  // 288 KB

    hipLaunchKernelGGL(dendrite_soma_kernel, grid, dim3(THREADS), shmem, stream,
                       x, idx, sw, sb, cwp, somab, out_soma, out_dend);
}